// SpatAttnLayer_77309411795
// MI455X (gfx1250) — compile-verified
//
#include <hip/hip_runtime.h>
#include <math.h>

#define B_  4
#define L_  12
#define N_  512
#define DM_ 64
#define H_  8
#define D_  8
#define SK_ 35

typedef float v2f __attribute__((ext_vector_type(2)));
typedef float v8f __attribute__((ext_vector_type(8)));
typedef unsigned int u32x4 __attribute__((ext_vector_type(4)));
typedef int i32x8 __attribute__((ext_vector_type(8)));
typedef int i32x4 __attribute__((ext_vector_type(4)));
typedef int v4i_vs __attribute__((vector_size(16)));   // matches builtin param type

#ifndef __has_builtin
#define __has_builtin(x) 0
#endif

#if __has_builtin(__builtin_amdgcn_tensor_load_to_lds) && \
    __has_builtin(__builtin_amdgcn_s_wait_tensorcnt)
#define HAVE_TDM 1
#else
#define HAVE_TDM 0
#endif

#if __has_builtin(__builtin_amdgcn_global_load_async_to_lds_b128) && \
    __has_builtin(__builtin_amdgcn_s_wait_asynccnt)
#define HAVE_ASYNC 1
#else
#define HAVE_ASYNC 0
#endif

// 16-byte global -> LDS stage: async (ASYNCcnt) when available, else plain copy.
__device__ __forceinline__ void stage16(const float* g, float* l) {
#if HAVE_ASYNC
    __builtin_amdgcn_global_load_async_to_lds_b128(
        (v4i_vs*)g,
        (__attribute__((address_space(3))) v4i_vs*)l, 0, 0);
#else
    *(float4*)l = *(const float4*)g;
#endif
}

__device__ __forceinline__ void stage_wait() {
#if HAVE_ASYNC
    __builtin_amdgcn_s_wait_asynccnt(0);
#endif
}

// ---------------------------------------------------------------------------
// 1) Regenerate jax idx table: threefry2x32 with key (0,42), bits & 511.
// ---------------------------------------------------------------------------
__device__ __forceinline__ unsigned rotl32(unsigned x, int r) {
    return (x << r) | (x >> (32 - r));
}

__global__ void idx_kernel(int* __restrict__ idx) {
    const int half = (N_ * SK_) / 2;   // 8960
    int i = blockIdx.x * blockDim.x + threadIdx.x;
    if (i >= half) return;
    const unsigned k0 = 0u, k1 = 42u;
    const unsigned k2 = k0 ^ k1 ^ 0x1BD11BDAu;
    unsigned ks[3] = {k0, k1, k2};
    unsigned x0 = (unsigned)i + k0;
    unsigned x1 = (unsigned)(i + half) + k1;
    const int rotA[4] = {13, 15, 26, 6};
    const int rotB[4] = {17, 29, 16, 24};
    #pragma unroll
    for (int g = 0; g < 5; ++g) {
        const int* rr = (g & 1) ? rotB : rotA;
        #pragma unroll
        for (int j = 0; j < 4; ++j) {
            x0 += x1;
            x1 = rotl32(x1, rr[j]);
            x1 ^= x0;
        }
        x0 += ks[(g + 1) % 3];
        x1 += ks[(g + 2) % 3] + (unsigned)(g + 1);
    }
    idx[i]        = (int)(x0 & 511u);
    idx[i + half] = (int)(x1 & 511u);
}

// ---------------------------------------------------------------------------
// 2) fp32 WMMA projection: Y(R,64) = X(R,64) @ W(64,64) + b, R = 24576.
//    W staged into LDS via the Tensor Data Mover (one wave issues the DMA).
// ---------------------------------------------------------------------------
__global__ __launch_bounds__(128) void proj_kernel(const float* __restrict__ X,
                                                   const float* __restrict__ W,
                                                   const float* __restrict__ bias,
                                                   float* __restrict__ Y) {
    __shared__ float sW[64 * 64];
    __shared__ float sB[64];
    const int tid = threadIdx.x;

#if HAVE_TDM
    if (tid < 32) {
        unsigned long long wa = (unsigned long long)(uintptr_t)W;
        unsigned ldsoff = (unsigned)(uintptr_t)sW;   // flat LDS addr low bits = LDS offset
        u32x4 g0;
        g0[0] = 1u;                                   // count=1, user descriptor
        g0[1] = ldsoff;                               // lds_addr
        g0[2] = (unsigned)(wa & 0xffffffffu);         // global_addr[31:0]
        g0[3] = (unsigned)((wa >> 32) & 0x01ffffffu)  // global_addr[56:32]
              | 0x80000000u;                          // type=2 ("image")
        i32x8 g1;
        g1[0] = 0x00020000;        // workgroup_mask=0, data_size=2 (4B)
        g1[1] = (int)(64u << 16);  // tensor_dim0 = 64 (bits 79:48 -> [31:16])
        g1[2] = (int)(64u << 16);  // tensor_dim1 = 64 (bits 111:80 -> [31:16])
        g1[3] = (int)(64u << 16);  // tile_dim0   = 64 (bits 127:112 -> [31:16])
        g1[4] = 64;                // tile_dim1 = 64, tile_dim2 = 0
        g1[5] = 64;                // tensor_dim0_stride = 64
        g1[6] = 0;
        g1[7] = 0;
        i32x4 gz4 = {0, 0, 0, 0};
        i32x8 gz8 = {0, 0, 0, 0, 0, 0, 0, 0};
        __builtin_amdgcn_tensor_load_to_lds(g0, g1, gz4, gz4, gz8, 0);
        __builtin_amdgcn_s_wait_tensorcnt(0);
    }
#else
    for (int i = tid; i < 64 * 64; i += 128) sW[i] = W[i];
#endif
    if (tid < 64) sB[tid] = bias[tid];
    __syncthreads();

    const int lane = tid & 31;
    const int wave = tid >> 5;
    const int rowBase = blockIdx.x * 64 + wave * 16;
    const int nl = lane & 15;
    const int hg = lane >> 4;

    v8f c[4];
    #pragma unroll
    for (int ct = 0; ct < 4; ++ct) {
        float bv = sB[ct * 16 + nl];
        #pragma unroll
        for (int r = 0; r < 8; ++r) c[ct][r] = bv;
    }

    const int arow = rowBase + nl;
    for (int kt = 0; kt < 16; ++kt) {
        const int k0 = kt * 4 + hg * 2;
        v2f a;
        a.x = X[arow * 64 + k0];
        a.y = X[arow * 64 + k0 + 1];
        #pragma unroll
        for (int ct = 0; ct < 4; ++ct) {
            v2f bf;
            bf.x = sW[k0 * 64 + ct * 16 + nl];
            bf.y = sW[(k0 + 1) * 64 + ct * 16 + nl];
            c[ct] = __builtin_amdgcn_wmma_f32_16x16x4_f32(
                false, a, false, bf, (short)0, c[ct], false, false);
        }
    }

    #pragma unroll
    for (int ct = 0; ct < 4; ++ct)
        #pragma unroll
        for (int r = 0; r < 8; ++r)
            Y[(rowBase + r + hg * 8) * 64 + ct * 16 + nl] = c[ct][r];
}

// ---------------------------------------------------------------------------
// 3) Sparsity measure, block per (b,h,l): K head-slice (512x8 = 16KB) staged
//    once into LDS via async-to-LDS, then 35 sampled dots per query row.
// ---------------------------------------------------------------------------
__global__ __launch_bounds__(256) void mscore_kernel(const float* __restrict__ Qp,
                                                     const float* __restrict__ Kp,
                                                     const int* __restrict__ idx,
                                                     float* __restrict__ Mout) {
    __shared__ float sK[N_ * D_];
    const int tid = threadIdx.x;
    const int g = blockIdx.x;
    const int l = g % L_;
    const int h = (g / L_) % H_;
    const int b = g / (L_ * H_);
    const size_t rowb = (size_t)(b * L_ + l) * N_;
    const float* kb = Kp + rowb * DM_ + h * D_;

    #pragma unroll
    for (int j = 0; j < 4; ++j) {
        int chunk = tid + 256 * j;      // 1024 chunks of 16B
        int m = chunk >> 1;
        int c = (chunk & 1) * 4;
        stage16(kb + (size_t)m * DM_ + c, sK + m * D_ + c);
    }
    stage_wait();
    __syncthreads();

    for (int n = tid; n < N_; n += 256) {
        __builtin_prefetch(idx + n * SK_, 0, 3);
        const float* qrow = Qp + (rowb + n) * DM_ + h * D_;
        float4 q0 = *(const float4*)qrow;
        float4 q1 = *(const float4*)(qrow + 4);
        float mx = -1e30f, sm = 0.f;
        for (int s = 0; s < SK_; ++s) {
            int m = idx[n * SK_ + s];
            const float* kr = sK + m * D_;
            float d = q0.x * kr[0] + q0.y * kr[1] + q0.z * kr[2] + q0.w * kr[3] +
                      q1.x * kr[4] + q1.y * kr[5] + q1.z * kr[6] + q1.w * kr[7];
            mx = fmaxf(mx, d);
            sm += d;
        }
        Mout[g * N_ + n] = mx - sm * (1.0f / N_);
    }
}

// ---------------------------------------------------------------------------
// 4) Per (b,h,l): top-35 indices of M (lowest-index tie-break like jax),
//    plus the V-mean over N for the default context rows.
// ---------------------------------------------------------------------------
__global__ __launch_bounds__(256) void topk_kernel(const float* __restrict__ Mv,
                                                   const float* __restrict__ Vp,
                                                   int* __restrict__ tops,
                                                   float* __restrict__ vmean) {
    __shared__ float vals[N_];
    __shared__ float rv[256];
    __shared__ int   ri[256];
    __shared__ float vp[256];
    const int tid = threadIdx.x;
    const int g = blockIdx.x;
    const int l = g % L_;
    const int h = (g / L_) % H_;
    const int b = g / (L_ * H_);

    vals[tid]       = Mv[g * N_ + tid];
    vals[tid + 256] = Mv[g * N_ + tid + 256];
    __syncthreads();

    for (int t = 0; t < SK_; ++t) {
        float bv = vals[tid];
        int bi = tid;
        float ov = vals[tid + 256];
        if (ov > bv) { bv = ov; bi = tid + 256; }
        rv[tid] = bv;
        ri[tid] = bi;
        __syncthreads();
        for (int s = 128; s > 0; s >>= 1) {
            if (tid < s) {
                float v2 = rv[tid + s];
                int   i2 = ri[tid + s];
                if (v2 > rv[tid] || (v2 == rv[tid] && i2 < ri[tid])) {
                    rv[tid] = v2;
                    ri[tid] = i2;
                }
            }
            __syncthreads();
        }
        if (tid == 0) {
            tops[g * SK_ + t] = ri[0];
            vals[ri[0]] = -1e30f;
        }
        __syncthreads();
    }

    const int d = tid & 7;
    const float* vb = Vp + (size_t)(b * L_ + l) * N_ * DM_ + h * D_ + d;
    float acc = 0.f;
    for (int n = tid >> 3; n < N_; n += 32) acc += vb[(size_t)n * DM_];
    vp[tid] = acc;
    __syncthreads();
    if (tid < 8) {
        float s = 0.f;
        for (int j = 0; j < 32; ++j) s += vp[(j << 3) | tid];
        vmean[g * 8 + tid] = s * (1.0f / N_);
    }
}

// ---------------------------------------------------------------------------
// 5) Default fill: out[b,l,n,h*8+d] = vmean[b,h,l,d].
// ---------------------------------------------------------------------------
__global__ void fill_kernel(const float* __restrict__ vmean,
                            float* __restrict__ out) {
    int t = blockIdx.x * blockDim.x + threadIdx.x;
    if (t >= B_ * L_ * N_ * DM_) return;
    int dm = t % DM_;
    int l = (t / (DM_ * N_)) % L_;
    int b = t / (DM_ * N_ * L_);
    int h = dm >> 3, d = dm & 7;
    out[t] = vmean[((b * H_ + h) * L_ + l) * 8 + d];
}

// ---------------------------------------------------------------------------
// 6) Full attention for the 35 selected rows per (b,h,l). K and V head-slices
//    (16KB each) staged once into LDS via async-to-LDS; softmax + weighted V
//    then run entirely out of LDS.
// ---------------------------------------------------------------------------
__global__ __launch_bounds__(256) void attn_kernel(const float* __restrict__ Qp,
                                                   const float* __restrict__ Kp,
                                                   const float* __restrict__ Vp,
                                                   const int* __restrict__ tops,
                                                   float* __restrict__ out) {
    __shared__ float sK[N_ * D_];
    __shared__ float sV[N_ * D_];
    __shared__ float sc[N_];
    __shared__ float red[256];
    __shared__ float vp[256];
    const int tid = threadIdx.x;
    const int g = blockIdx.x;
    const int l = g % L_;
    const int h = (g / L_) % H_;
    const int b = g / (L_ * H_);
    const size_t rowb = (size_t)(b * L_ + l) * N_;
    const float scale = 0.3535533905932738f; // 1/sqrt(D)

    const float* kb = Kp + rowb * DM_ + h * D_;
    const float* vb = Vp + rowb * DM_ + h * D_;
    #pragma unroll
    for (int j = 0; j < 4; ++j) {
        int chunk = tid + 256 * j;
        int m = chunk >> 1;
        int c = (chunk & 1) * 4;
        stage16(kb + (size_t)m * DM_ + c, sK + m * D_ + c);
        stage16(vb + (size_t)m * DM_ + c, sV + m * D_ + c);
    }
    stage_wait();
    __syncthreads();

    for (int t = 0; t < SK_; ++t) {
        const int nq = tops[g * SK_ + t];
        const float* qrow = Qp + (rowb + nq) * DM_ + h * D_;
        float4 q0 = *(const float4*)qrow;
        float4 q1 = *(const float4*)(qrow + 4);

        for (int m = tid; m < N_; m += 256) {
            const float* kr = sK + m * D_;
            float d = q0.x * kr[0] + q0.y * kr[1] + q0.z * kr[2] + q0.w * kr[3] +
                      q1.x * kr[4] + q1.y * kr[5] + q1.z * kr[6] + q1.w * kr[7];
            sc[m] = d * scale;
        }
        __syncthreads();

        red[tid] = fmaxf(sc[tid], sc[tid + 256]);
        __syncthreads();
        for (int s = 128; s > 0; s >>= 1) {
            if (tid < s) red[tid] = fmaxf(red[tid], red[tid + s]);
            __syncthreads();
        }
        float mx = red[0];
        __syncthreads();

        float e0 = __expf(sc[tid] - mx);
        float e1 = __expf(sc[tid + 256] - mx);
        sc[tid] = e0;
        sc[tid + 256] = e1;
        red[tid] = e0 + e1;
        __syncthreads();
        for (int s = 128; s > 0; s >>= 1) {
            if (tid < s) red[tid] += red[tid + s];
            __syncthreads();
        }
        float denom = red[0];

        const int d = tid & 7;
        float acc = 0.f;
        for (int m = tid >> 3; m < N_; m += 32) acc += sc[m] * sV[m * D_ + d];
        vp[tid] = acc;
        __syncthreads();
        if (tid < 8) {
            float s2 = 0.f;
            for (int j = 0; j < 32; ++j) s2 += vp[(j << 3) | tid];
            out[(rowb + nq) * DM_ + h * D_ + tid] = s2 / denom;
        }
        __syncthreads();
    }
}

// ---------------------------------------------------------------------------
extern "C" void kernel_launch(void* const* d_in, const int* in_sizes, int n_in,
                              void* d_out, int out_size, void* d_ws, size_t ws_size,
                              hipStream_t stream) {
    const float* queries = (const float*)d_in[0];
    const float* keys    = (const float*)d_in[1];
    const float* values  = (const float*)d_in[2];
    const float* Wq      = (const float*)d_in[3];
    const float* bq      = (const float*)d_in[4];
    const float* Wk      = (const float*)d_in[5];
    const float* bk      = (const float*)d_in[6];
    const float* Wv      = (const float*)d_in[7];
    const float* bv      = (const float*)d_in[8];
    float* out = (float*)d_out;

    char* ws = (char*)d_ws;
    size_t off = 0;
    auto alloc = [&](size_t bytes) -> void* {
        void* p = ws + off;
        off += (bytes + 255) & ~(size_t)255;
        return p;
    };
    const size_t projN = (size_t)B_ * L_ * N_ * DM_;
    float* Qp    = (float*)alloc(sizeof(float) * projN);
    float* Kp    = (float*)alloc(sizeof(float) * projN);
    float* Vp    = (float*)alloc(sizeof(float) * projN);
    float* Mv    = (float*)alloc(sizeof(float) * B_ * H_ * L_ * N_);
    float* vmean = (float*)alloc(sizeof(float) * B_ * H_ * L_ * D_);
    int*   idx   = (int*)alloc(sizeof(int) * N_ * SK_);
    int*   tops  = (int*)alloc(sizeof(int) * B_ * H_ * L_ * SK_);
    (void)ws_size;

    const int R = B_ * L_ * N_;             // 24576 rows
    idx_kernel<<<(N_ * SK_ / 2 + 255) / 256, 256, 0, stream>>>(idx);
    proj_kernel<<<R / 64, 128, 0, stream>>>(queries, Wq, bq, Qp);
    proj_kernel<<<R / 64, 128, 0, stream>>>(keys,    Wk, bk, Kp);
    proj_kernel<<<R / 64, 128, 0, stream>>>(values,  Wv, bv, Vp);
    mscore_kernel<<<B_ * H_ * L_, 256, 0, stream>>>(Qp, Kp, idx, Mv);
    topk_kernel<<<B_ * H_ * L_, 256, 0, stream>>>(Mv, Vp, tops, vmean);
    fill_kernel<<<(int)((projN + 255) / 256), 256, 0, stream>>>(vmean, out);
    attn_kernel<<<B_ * H_ * L_, 256, 0, stream>>>(Qp, Kp, Vp, tops, out);
}